// MultiheadAttention_9105330667631
// MI455X (gfx1250) — compile-verified
//
#include <hip/hip_runtime.h>
#include <math.h>
#include <stdint.h>

// ---------------- problem constants (match reference) ----------------
#define BB 16
#define SS 1024
#define DD 512
#define HH 8
#define HD 64
#define ROWS (BB*SS)                  // 16384
#define QKV_ELEMS ((size_t)ROWS*DD)   // 8,388,608 floats per tensor

typedef float v2f __attribute__((ext_vector_type(2)));
typedef float v8f __attribute__((ext_vector_type(8)));

static __device__ __forceinline__ v8f wmma_f32(v2f a, v2f b, v8f c) {
    // V_WMMA_F32_16X16X4_F32 : D(16x16 f32) = A(16x4 f32) * B(4x16 f32) + C
    return __builtin_amdgcn_wmma_f32_16x16x4_f32(
        /*neg_a=*/false, a, /*neg_b=*/false, b,
        /*c_mod=*/(short)0, c, /*reuse_a=*/false, /*reuse_b=*/false);
}

// ---------------------------------------------------------------------
// Kernel 1: row validity mask  valid[b,s] = any(Q[b,s,:] != 0)
// wave-per-row, lane covers 16 contiguous floats
// ---------------------------------------------------------------------
__global__ void valid_kernel(const float* __restrict__ Q, int* __restrict__ valid) {
    const int wave = threadIdx.x >> 5;
    const int lane = threadIdx.x & 31;
    const int row  = blockIdx.x * 8 + wave;
    const float* p = Q + (size_t)row * DD + lane * 16;
    int nz = 0;
#pragma unroll
    for (int i = 0; i < 4; ++i) {
        float4 t = *(const float4*)(p + i * 4);
        nz |= (t.x != 0.f) | (t.y != 0.f) | (t.z != 0.f) | (t.w != 0.f);
    }
    for (int off = 16; off > 0; off >>= 1) nz |= __shfl_xor(nz, off, 32);
    if (lane == 0) valid[row] = nz;
}

// ---------------------------------------------------------------------
// Kernel 2: fused QKV projection  X = relu(Q @ W^T + b)
// grid = (ROWS/16, 1, 3)   block = 256 (8 waves)
// wave w computes a 16(M) x 64(N) tile: A fragment reused over 4 N-subtiles
// NT layout: per lane both fragments are contiguous float2 along K.
// ---------------------------------------------------------------------
__global__ void qkv_proj(const float* __restrict__ Q,
                         const float* __restrict__ Wq, const float* __restrict__ bq,
                         const float* __restrict__ Wk, const float* __restrict__ bk,
                         const float* __restrict__ Wv, const float* __restrict__ bv,
                         float* __restrict__ oq, float* __restrict__ ok,
                         float* __restrict__ ov) {
    const int z = blockIdx.z;
    const float* W    = (z == 0) ? Wq : ((z == 1) ? Wk : Wv);
    const float* bias = (z == 0) ? bq : ((z == 1) ? bk : bv);
    float*       out  = (z == 0) ? oq : ((z == 1) ? ok : ov);

    const int tid  = threadIdx.x;
    const int wave = tid >> 5;
    const int lane = tid & 31;
    const int lm   = lane & 15;     // M (A) or N (B) index within 16
    const int hi   = lane >> 4;     // K sub-pair selector
    const int m0   = blockIdx.x * 16;
    const int n0w  = wave * 64;

    v8f acc0 = {}, acc1 = {}, acc2 = {}, acc3 = {};
    const float* arow = Q + (size_t)(m0 + lm) * DD + 2 * hi;
    const float* b0 = W + (size_t)(n0w +  0 + lm) * DD + 2 * hi;
    const float* b1 = W + (size_t)(n0w + 16 + lm) * DD + 2 * hi;
    const float* b2 = W + (size_t)(n0w + 32 + lm) * DD + 2 * hi;
    const float* b3 = W + (size_t)(n0w + 48 + lm) * DD + 2 * hi;

    for (int kk = 0; kk < DD; kk += 4) {
        v2f a  = *(const v2f*)(arow + kk);
        v2f f0 = *(const v2f*)(b0 + kk);
        v2f f1 = *(const v2f*)(b1 + kk);
        v2f f2 = *(const v2f*)(b2 + kk);
        v2f f3 = *(const v2f*)(b3 + kk);
        acc0 = wmma_f32(a, f0, acc0);
        acc1 = wmma_f32(a, f1, acc1);
        acc2 = wmma_f32(a, f2, acc2);
        acc3 = wmma_f32(a, f3, acc3);
    }
    // C/D layout: lanes 0-15 N=lm (M = j), lanes 16-31 N=lm (M = 8+j)
#pragma unroll
    for (int j = 0; j < 8; ++j) {
        const int m = m0 + j + 8 * hi;
        const int c0 = n0w +  0 + lm, c1 = n0w + 16 + lm;
        const int c2 = n0w + 32 + lm, c3 = n0w + 48 + lm;
        out[(size_t)m * DD + c0] = fmaxf(acc0[j] + bias[c0], 0.f);
        out[(size_t)m * DD + c1] = fmaxf(acc1[j] + bias[c1], 0.f);
        out[(size_t)m * DD + c2] = fmaxf(acc2[j] + bias[c2], 0.f);
        out[(size_t)m * DD + c3] = fmaxf(acc3[j] + bias[c3], 0.f);
    }
}

// ---------------------------------------------------------------------
// Kernel 3: fused attention per (b,h, 16-row q strip)
// grid = (S/16, B*H)  block = 256 (8 waves)
// dynamic LDS: sS[16*1024] | sQ[16*64] | sPart[8*256] | sQv[16]
// q tile staged via GLOBAL_LOAD_ASYNC_TO_LDS_B128 (ASYNCcnt path)
// normalized weights streamed to HBM with non-temporal stores
// ---------------------------------------------------------------------
__global__ void attn_kernel(const float* __restrict__ q,
                            const float* __restrict__ k,
                            const float* __restrict__ v,
                            const int*   __restrict__ valid,
                            float* __restrict__ outPre,   // [B,S,D] pre-LN
                            float* __restrict__ wOut) {   // [B*H,S,S]
    extern __shared__ float smem[];
    float* sS    = smem;            // 16 x 1024 scores/weights
    float* sQ    = smem + 16384;    // 16 x 64 q tile
    float* sPart = smem + 17408;    // 8 x 256 PV partials
    int*   sQv   = (int*)(smem + 19456);

    const int qblk = blockIdx.x;          // 0..63
    const int bh   = blockIdx.y;          // 0..127
    const int b    = bh >> 3, h = bh & 7;
    const int qr0  = qblk * 16;
    const int tid  = threadIdx.x, wave = tid >> 5, lane = tid & 31;
    const int lm   = lane & 15, hi = lane >> 4;
    const size_t baseBH = (size_t)b * SS * DD + h * HD;   // + s*DD + d

    // ---- async DMA q tile [16,64] into LDS: 16B per lane, 4KB total ----
    {
        const int idx = tid * 4;
        const int r = idx >> 6, c = idx & 63;
        const float* gsrc = q + baseBH + (size_t)(qr0 + r) * DD + c;
        const unsigned lds_dst = (unsigned)(uintptr_t)(&sQ[idx]); // LDS aperture: addr[31:0]
        asm volatile("global_load_async_to_lds_b128 %0, %1, off"
                     :: "v"(lds_dst), "v"(gsrc) : "memory");
    }
    if (tid < 16) sQv[tid] = valid[b * SS + qr0 + tid];
    asm volatile("s_wait_asynccnt 0" ::: "memory");
    __syncthreads();

    // ---- scores: wave covers 128 key columns (8 tiles of 16), K = 64 ----
    const float scale = 0.125f;  // 1/sqrt(64)
    for (int t = 0; t < 8; ++t) {
        const int n0 = wave * 128 + t * 16;
        v8f acc = {};
        const float* arow = sQ + lm * HD + 2 * hi;
        const float* brow = k + baseBH + (size_t)(n0 + lm) * DD + 2 * hi;
#pragma unroll
        for (int kk = 0; kk < HD; kk += 4) {
            v2f a  = *(const v2f*)(arow + kk);
            v2f bb = *(const v2f*)(brow + kk);
            acc = wmma_f32(a, bb, acc);
        }
        const int kcol = n0 + lm;
        const int kval = valid[b * SS + kcol];
#pragma unroll
        for (int j = 0; j < 8; ++j) {
            const int m = j + 8 * hi;
            float s = acc[j] * scale;
            s = (sQv[m] && kval) ? s : -INFINITY;
            sS[m * SS + kcol] = s;
        }
    }
    __syncthreads();

    // ---- softmax per row (wave handles 2 rows); fully-masked rows -> 0 ----
    for (int rr = 0; rr < 2; ++rr) {
        const int row = wave * 2 + rr;
        float* srow = sS + row * SS;
        float mx = -INFINITY;
#pragma unroll 4
        for (int i = 0; i < 32; ++i) mx = fmaxf(mx, srow[lane + i * 32]);
        for (int off = 16; off > 0; off >>= 1) mx = fmaxf(mx, __shfl_xor(mx, off, 32));
        const bool dead = (mx == -INFINITY);
        float sum = 0.f;
#pragma unroll 4
        for (int i = 0; i < 32; ++i) {
            const int c = lane + i * 32;
            float e = dead ? 0.f : __expf(srow[c] - mx);
            srow[c] = e;
            sum += e;
        }
        for (int off = 16; off > 0; off >>= 1) sum += __shfl_xor(sum, off, 32);
        const float inv = (dead || sum <= 0.f) ? 0.f : 1.f / sum;
        float* wrow = wOut + ((size_t)bh * SS + qr0 + row) * SS;
#pragma unroll 4
        for (int i = 0; i < 32; ++i) {
            const int c = lane + i * 32;
            const float wv_ = srow[c] * inv;
            srow[c] = wv_;
            __builtin_nontemporal_store(wv_, &wrow[c]);  // 536MB write-once stream: keep L2 for q/k/v
        }
    }
    __syncthreads();

    // ---- P@V: wave = (khalf<<2)|tile ; tile = 16 cols, khalf = 512 keys ----
    {
        const int tile = wave & 3, kh = wave >> 2;
        const int n0 = tile * 16;
        v8f acc = {};
        const float* arow  = sS + lm * SS + 2 * hi;
        const float* vbase = v + baseBH + n0 + lm + (size_t)(2 * hi) * DD;
        for (int kk = kh * 512; kk < kh * 512 + 512; kk += 4) {
            v2f a = *(const v2f*)(arow + kk);
            const float* vb = vbase + (size_t)kk * DD;
            v2f bb;
            bb.x = vb[0];      // V[kk+2*hi,   n0+lm]
            bb.y = vb[DD];     // V[kk+2*hi+1, n0+lm]
            acc = wmma_f32(a, bb, acc);
        }
#pragma unroll
        for (int j = 0; j < 8; ++j)
            sPart[wave * 256 + (j + 8 * hi) * 16 + lm] = acc[j];
    }
    __syncthreads();

    // ---- combine K-halves + residual q, write pre-LN output ----
#pragma unroll
    for (int e = 0; e < 4; ++e) {
        const int idx = tid + e * 256;
        const int m = idx >> 6, n = idx & 63;
        const int tile = n >> 4, nn = n & 15;
        const float val = sPart[tile * 256 + m * 16 + nn]
                        + sPart[(tile + 4) * 256 + m * 16 + nn]
                        + sQ[m * HD + n];
        outPre[baseBH + (size_t)(qr0 + m) * DD + n] = val;
    }
}

// ---------------------------------------------------------------------
// Kernel 4: in-place LayerNorm (ddof=0) + ReLU over rows of d_out
// wave-per-row, 8 rows per block
// ---------------------------------------------------------------------
__global__ void ln_kernel(float* __restrict__ out,
                          const float* __restrict__ g,
                          const float* __restrict__ beta) {
    const int wave = threadIdx.x >> 5, lane = threadIdx.x & 31;
    const int row = blockIdx.x * 8 + wave;
    float* p = out + (size_t)row * DD;
    float xv[16];
    float sum = 0.f, sq = 0.f;
#pragma unroll
    for (int i = 0; i < 16; ++i) {
        const float x = p[lane + i * 32];
        xv[i] = x; sum += x; sq += x * x;
    }
    for (int off = 16; off > 0; off >>= 1) {
        sum += __shfl_xor(sum, off, 32);
        sq  += __shfl_xor(sq,  off, 32);
    }
    const float mu  = sum * (1.f / DD);
    const float var = fmaxf(sq * (1.f / DD) - mu * mu, 0.f);
    const float rs  = rsqrtf(var + 1e-5f);
#pragma unroll
    for (int i = 0; i < 16; ++i) {
        const int c = lane + i * 32;
        p[c] = fmaxf((xv[i] - mu) * rs * g[c] + beta[c], 0.f);
    }
}

// ---------------------------------------------------------------------
extern "C" void kernel_launch(void* const* d_in, const int* in_sizes, int n_in,
                              void* d_out, int out_size, void* d_ws, size_t ws_size,
                              hipStream_t stream) {
    const float* Q    = (const float*)d_in[0];
    const float* Wq_w = (const float*)d_in[1];
    const float* Wq_b = (const float*)d_in[2];
    const float* Wk_w = (const float*)d_in[3];
    const float* Wk_b = (const float*)d_in[4];
    const float* Wv_w = (const float*)d_in[5];
    const float* Wv_b = (const float*)d_in[6];
    const float* ln_w = (const float*)d_in[7];
    const float* ln_b = (const float*)d_in[8];

    // workspace: q | k | v | valid
    float* qbuf = (float*)d_ws;
    float* kbuf = qbuf + QKV_ELEMS;
    float* vbuf = kbuf + QKV_ELEMS;
    int*   valid = (int*)(vbuf + QKV_ELEMS);

    // d_out: out [B,S,D] then weights [B*H,S,S]
    float* outPre = (float*)d_out;
    float* wOut   = outPre + QKV_ELEMS;

    valid_kernel<<<ROWS / 8, 256, 0, stream>>>(Q, valid);

    qkv_proj<<<dim3(ROWS / 16, 1, 3), 256, 0, stream>>>(
        Q, Wq_w, Wq_b, Wk_w, Wk_b, Wv_w, Wv_b, qbuf, kbuf, vbuf);

    const size_t smem = (16 * 1024 + 16 * 64 + 8 * 256 + 16) * sizeof(float); // 77,888 B
    attn_kernel<<<dim3(SS / 16, BB * HH), 256, smem, stream>>>(
        qbuf, kbuf, vbuf, valid, outPre, wOut);

    ln_kernel<<<ROWS / 8, 256, 0, stream>>>(outPre, ln_w, ln_b);
}